// KernelLayer_65481071403945
// MI455X (gfx1250) — compile-verified
//
#include <hip/hip_runtime.h>

typedef __attribute__((ext_vector_type(2))) float v2f;
typedef __attribute__((ext_vector_type(8))) float v8f;

#define GAMMA_F   1.0f
#define D_DIM     64
#define KAUG      68          // 64 data + ||x||^2 + 1 + 2 pad ; 68 % 64 == 4 -> conflict-free LDS
#define TILE_N    128
#define TILE_M    64

__global__ __launch_bounds__(256)
void rbf_wmma_f32_kernel(const float* __restrict__ x,
                         const float* __restrict__ p,
                         float* __restrict__ out,
                         int N, int M) {
    __shared__ float Alds[TILE_N * KAUG];   // augmented x panel   [row][k]
    __shared__ float Blds[TILE_M * KAUG];   // augmented -2p panel [col][k]

    const int tid  = threadIdx.x;
    const int row0 = blockIdx.y * TILE_N;
    const int col0 = blockIdx.x * TILE_M;

    // ---- Stage x tile (128 x 64 fp32) into LDS, float4 moves --------------
    {
        const float4* xg = (const float4*)(x + (size_t)row0 * D_DIM);
        #pragma unroll
        for (int i = 0; i < 8; ++i) {
            int idx = tid + 256 * i;      // 0..2047 float4s
            int r   = idx >> 4;           // row within tile
            int c4  = idx & 15;           // float4 column
            float4 v = xg[r * 16 + c4];
            *(float4*)&Alds[r * KAUG + c4 * 4] = v;
        }
    }
    // ---- Stage prototype tile (64 x 64), pre-scaled by -2 -----------------
    {
        const float4* pg = (const float4*)(p + (size_t)col0 * D_DIM);
        #pragma unroll
        for (int i = 0; i < 4; ++i) {
            int idx = tid + 256 * i;      // 0..1023 float4s
            int r   = idx >> 4;
            int c4  = idx & 15;
            float4 v = pg[r * 16 + c4];
            float4 w;
            w.x = -2.0f * v.x; w.y = -2.0f * v.y;
            w.z = -2.0f * v.z; w.w = -2.0f * v.w;
            *(float4*)&Blds[r * KAUG + c4 * 4] = w;
        }
    }
    __syncthreads();

    // ---- Augmentation columns: A'=[x,|x|^2,1,0,0]  B'=[-2p,1,|p|^2,0,0] ---
    if (tid < TILE_N) {
        float s = 0.0f;
        #pragma unroll 16
        for (int k = 0; k < D_DIM; ++k) { float v = Alds[tid * KAUG + k]; s += v * v; }
        Alds[tid * KAUG + 64] = s;        // ||x||^2
        Alds[tid * KAUG + 65] = 1.0f;
        Alds[tid * KAUG + 66] = 0.0f;
        Alds[tid * KAUG + 67] = 0.0f;
    } else if (tid < TILE_N + TILE_M) {
        int r = tid - TILE_N;
        float s = 0.0f;
        #pragma unroll 16
        for (int k = 0; k < D_DIM; ++k) { float v = Blds[r * KAUG + k]; s += v * v; }
        Blds[r * KAUG + 64] = 1.0f;
        Blds[r * KAUG + 65] = 0.25f * s;  // ||p||^2 recovered from (-2p)^2
        Blds[r * KAUG + 66] = 0.0f;
        Blds[r * KAUG + 67] = 0.0f;
    }
    __syncthreads();

    // ---- WMMA main loop: each wave -> 16 rows x 64 cols (4 accumulators) --
    const int wave  = tid >> 5;
    const int lane  = tid & 31;
    const int lhalf = lane >> 4;          // 0: k+{0,1} | 1: k+{2,3}  (f32 16x16x4 layout)
    const int l16   = lane & 15;

    v8f acc0 = {}, acc1 = {}, acc2 = {}, acc3 = {};

    const int aoff  = (wave * 16 + l16) * KAUG + 2 * lhalf;
    const int boff0 = ( 0 + l16) * KAUG + 2 * lhalf;
    const int boff1 = (16 + l16) * KAUG + 2 * lhalf;
    const int boff2 = (32 + l16) * KAUG + 2 * lhalf;
    const int boff3 = (48 + l16) * KAUG + 2 * lhalf;

    #pragma unroll
    for (int kc = 0; kc < KAUG; kc += 4) {
        v2f a  = *(const v2f*)&Alds[aoff  + kc];
        v2f b0 = *(const v2f*)&Blds[boff0 + kc];
        v2f b1 = *(const v2f*)&Blds[boff1 + kc];
        v2f b2 = *(const v2f*)&Blds[boff2 + kc];
        v2f b3 = *(const v2f*)&Blds[boff3 + kc];
        acc0 = __builtin_amdgcn_wmma_f32_16x16x4_f32(false, a, false, b0, (short)0, acc0, false, false);
        acc1 = __builtin_amdgcn_wmma_f32_16x16x4_f32(false, a, false, b1, (short)0, acc1, false, false);
        acc2 = __builtin_amdgcn_wmma_f32_16x16x4_f32(false, a, false, b2, (short)0, acc2, false, false);
        acc3 = __builtin_amdgcn_wmma_f32_16x16x4_f32(false, a, false, b3, (short)0, acc3, false, false);
    }

    // ---- Epilogue: dist2 -> exp(-gamma * max(dist2,0)), direct stores -----
    // C/D layout: lane l -> N=l&15 ; VGPR v -> M = v + 8*(l>>4)
    float* orow = out + (size_t)(row0 + wave * 16) * M + col0;
    #pragma unroll
    for (int v = 0; v < 8; ++v) {
        const size_t rbase = (size_t)(v + 8 * lhalf) * M;
        float d0 = fmaxf(acc0[v], 0.0f);
        float d1 = fmaxf(acc1[v], 0.0f);
        float d2 = fmaxf(acc2[v], 0.0f);
        float d3 = fmaxf(acc3[v], 0.0f);
        orow[rbase +  0 + l16] = __expf(-GAMMA_F * d0);
        orow[rbase + 16 + l16] = __expf(-GAMMA_F * d1);
        orow[rbase + 32 + l16] = __expf(-GAMMA_F * d2);
        orow[rbase + 48 + l16] = __expf(-GAMMA_F * d3);
    }
}

extern "C" void kernel_launch(void* const* d_in, const int* in_sizes, int n_in,
                              void* d_out, int out_size, void* d_ws, size_t ws_size,
                              hipStream_t stream) {
    const float* x = (const float*)d_in[0];       // [N, 64] fp32
    const float* p = (const float*)d_in[1];       // [M, 64] fp32
    float* out = (float*)d_out;                   // [N, M]  fp32

    const int N = in_sizes[0] / D_DIM;            // 131072
    const int M = in_sizes[1] / D_DIM;            // 512

    dim3 grid((unsigned)(M / TILE_M), (unsigned)(N / TILE_N));
    rbf_wmma_f32_kernel<<<grid, 256, 0, stream>>>(x, p, out, N, M);
}